// SoftDiceLoss_15487652069653
// MI455X (gfx1250) — compile-verified
//
#include <hip/hip_runtime.h>

// Soft Dice loss over 26-class int32 label maps (2x192x192x192 each).
// Stage 1: streaming histogram (b128 loads, per-wave LDS histograms, ds_add_u32),
//          per-block 96-entry f32 partials to d_ws.
// Stage 2: one block; 6 waves reduce all block partials with
//          v_wmma_f32_16x16x4_f32 (A = ones), then lane math for the dice mean.

#define NUM_CLASSES 26
#define EPSILONF    1e-5f

#define H_FIX 0
#define H_MOV 32
#define H_INT 64
#define HIST_STRIDE 96   // 3 histograms x 32 padded classes
#define TPB 256
#define WAVES_PER_BLOCK (TPB / 32)
#define MAX_BLOCKS 1024

typedef __attribute__((ext_vector_type(2))) float v2f;
typedef __attribute__((ext_vector_type(8))) float v8f;

__global__ __launch_bounds__(TPB)
void SoftDice_hist_kernel(const int* __restrict__ f, const int* __restrict__ m,
                          float* __restrict__ partials, int n, int n4) {
    __shared__ int h[WAVES_PER_BLOCK][HIST_STRIDE];

    const int wave = threadIdx.x >> 5;

    // Zero the per-wave private histograms.
    for (int i = threadIdx.x; i < WAVES_PER_BLOCK * HIST_STRIDE; i += TPB)
        (&h[0][0])[i] = 0;
    __syncthreads();

    int* hw = h[wave];

    const int4* __restrict__ f4 = (const int4*)f;
    const int4* __restrict__ m4 = (const int4*)m;
    const int stride = gridDim.x * TPB;

    for (int i = blockIdx.x * TPB + threadIdx.x; i < n4; i += stride) {
        if (i + stride < n4) {
            // CDNA5 global_prefetch_b8: pull next grid-stride chunk toward L2/WGP$.
            __builtin_prefetch(&f4[i + stride], 0, 1);
            __builtin_prefetch(&m4[i + stride], 0, 1);
        }
        int4 a = f4[i];     // global_load_b128
        int4 b = m4[i];

        // &31 masks keep LDS in-bounds for any input; labels are 0..25.
        atomicAdd(&hw[H_FIX + (a.x & 31)], 1);
        atomicAdd(&hw[H_FIX + (a.y & 31)], 1);
        atomicAdd(&hw[H_FIX + (a.z & 31)], 1);
        atomicAdd(&hw[H_FIX + (a.w & 31)], 1);

        atomicAdd(&hw[H_MOV + (b.x & 31)], 1);
        atomicAdd(&hw[H_MOV + (b.y & 31)], 1);
        atomicAdd(&hw[H_MOV + (b.z & 31)], 1);
        atomicAdd(&hw[H_MOV + (b.w & 31)], 1);

        if (a.x == b.x) atomicAdd(&hw[H_INT + (a.x & 31)], 1);
        if (a.y == b.y) atomicAdd(&hw[H_INT + (a.y & 31)], 1);
        if (a.z == b.z) atomicAdd(&hw[H_INT + (a.z & 31)], 1);
        if (a.w == b.w) atomicAdd(&hw[H_INT + (a.w & 31)], 1);
    }

    // Tail (n not divisible by 4): at most 3 elements, handled by block 0.
    {
        const int gtid = blockIdx.x * TPB + threadIdx.x;
        const int tail = n - n4 * 4;
        if (gtid < tail) {
            int fa = f[n4 * 4 + gtid] & 31;
            int mb = m[n4 * 4 + gtid] & 31;
            atomicAdd(&hw[H_FIX + fa], 1);
            atomicAdd(&hw[H_MOV + mb], 1);
            if (fa == mb) atomicAdd(&hw[H_INT + fa], 1);
        }
    }
    __syncthreads();

    // Reduce the 8 wave-private histograms, emit f32 partials (exact: counts < 2^24).
    for (int j = threadIdx.x; j < HIST_STRIDE; j += TPB) {
        int s = 0;
        for (int w = 0; w < WAVES_PER_BLOCK; ++w) s += h[w][j];
        partials[blockIdx.x * HIST_STRIDE + j] = (float)s;
    }
}

__global__ __launch_bounds__(256)
void SoftDice_reduce_kernel(const float* __restrict__ partials,
                            float* __restrict__ out, int nblocks) {
    __shared__ float totals[HIST_STRIDE];

    const int wave = threadIdx.x >> 5;
    const int lane = threadIdx.x & 31;

    if (wave < 6) {  // wave-uniform branch: EXEC all-ones inside, as WMMA requires
        // Column j = wave*16 + (lane%16) of the 96 histogram counters.
        const int col  = wave * 16 + (lane & 15);
        const int half = lane >> 4;  // which lane-half -> which K-row slot

        // A = 16x4 ones: any A-layout yields column sums of B in every D row.
        v2f a; a[0] = 1.0f; a[1] = 1.0f;
        v8f acc = {};

        // B = 4 consecutive block-partials x 16 columns. Summation over K is
        // symmetric, so the K-slot <-> (lane-half, VGPR) assignment is irrelevant;
        // each of blocks b..b+3 occupies exactly one of the 4 slots per column.
        for (int b = 0; b < nblocks; b += 4) {
            v2f bm;
            bm[0] = partials[(b + half    ) * HIST_STRIDE + col];
            bm[1] = partials[(b + half + 2) * HIST_STRIDE + col];
            acc = __builtin_amdgcn_wmma_f32_16x16x4_f32(
                /*neg_a=*/false, a, /*neg_b=*/false, bm,
                /*c_mod=*/(short)0, acc, /*reuse_a=*/false, /*reuse_b=*/false);
        }

        // Documented C/D layout: VGPR0, lanes 0-15 hold row M=0, N=lane.
        if (lane < 16) totals[col] = acc[0];
    }
    __syncthreads();

    if (threadIdx.x == 0) {
        float s = 0.0f;
        for (int c = 1; c < NUM_CLASSES; ++c) {
            float F = totals[H_FIX + c];
            float M = totals[H_MOV + c];
            float I = totals[H_INT + c];
            s += (2.0f * I + EPSILONF) / (F + M + EPSILONF);
        }
        out[0] = 1.0f - s * (1.0f / (NUM_CLASSES - 1));
    }
}

extern "C" void kernel_launch(void* const* d_in, const int* in_sizes, int n_in,
                              void* d_out, int out_size, void* d_ws, size_t ws_size,
                              hipStream_t stream) {
    const int* f = (const int*)d_in[0];
    const int* m = (const int*)d_in[1];
    float* out      = (float*)d_out;
    float* partials = (float*)d_ws;

    const int n  = in_sizes[0];
    const int n4 = n / 4;

    // Deterministic block count: MAX_BLOCKS, clamped to workspace, multiple of 4.
    int blocks = MAX_BLOCKS;
    int maxByWs = (int)(ws_size / (HIST_STRIDE * sizeof(float))) & ~3;
    if (blocks > maxByWs) blocks = maxByWs;
    if (blocks < 4) blocks = 4;

    SoftDice_hist_kernel<<<blocks, TPB, 0, stream>>>(f, m, partials, n, n4);
    SoftDice_reduce_kernel<<<1, 256, 0, stream>>>(partials, out, blocks);
}